// GCNConv_59785944760968
// MI455X (gfx1250) — compile-verified
//
#include <hip/hip_runtime.h>
#include <hip/hip_bf16.h>

typedef __attribute__((ext_vector_type(2))) float v2f;
typedef __attribute__((ext_vector_type(8))) float v8f;

#define D_FEAT 128
#define LDS_STRIDE 132   // pad 128 -> 132 dwords: A-frag ds_load_b64 hits all 64 banks

// ---------------- kernel 1: zero accumulators ----------------
__global__ void gcn_zero_kernel(float* __restrict__ p, int n) {
    int i = blockIdx.x * blockDim.x + threadIdx.x;
    if (i < n) p[i] = 0.0f;
}

// ---------------- kernel 2: gather + atomic segment-sum ----------------
// one wave32 per edge: lane loads float4 of x[col], atomically adds into sums[row]
__global__ __launch_bounds__(256) void gcn_scatter_kernel(
        const float* __restrict__ x,
        const long long* __restrict__ ei,   // [2, E] int64: row=ei[e], col=ei[E+e]
        float* __restrict__ sums,
        float* __restrict__ cnts,
        int E) {
    int g = blockIdx.x * blockDim.x + threadIdx.x;
    int e = g >> 5;
    if (e >= E) return;
    int lane = g & 31;

    int row = (int)ei[e];
    int col = (int)ei[(size_t)E + e];

    const float4* src = (const float4*)(x + (size_t)col * D_FEAT);
    float4 v = src[lane];                       // global_load_b128, coalesced per wave

    float* dst = sums + (size_t)row * D_FEAT + lane * 4;
    atomicAdd(dst + 0, v.x);                    // global_atomic_add_f32 (L2 atomics)
    atomicAdd(dst + 1, v.y);
    atomicAdd(dst + 2, v.z);
    atomicAdd(dst + 3, v.w);
    if (lane == 0) atomicAdd(cnts + row, 1.0f);
}

// ---------------- kernel 3: fused mean-normalize + WMMA GEMM + bias ----------------
// out[m][n] = mean[m][k] * W[n][k] + b[n]   (y = mean @ W^T + b)
// block = 16 output rows x 128 cols; 8 waves, one 16x16 N-tile each; K=128 via 32x WMMA k4
__global__ __launch_bounds__(256) void gcn_mean_gemm_kernel(
        const float* __restrict__ sums,
        const float* __restrict__ cnts,
        const float* __restrict__ W,       // [128,128] row-major: W[n][k]
        const float* __restrict__ bias,    // [128]
        float* __restrict__ out) {
    __shared__ float ldsA[16 * LDS_STRIDE];

    const int row0 = blockIdx.x * 16;
    const int tid  = threadIdx.x;

    // normalize 16x128 tile into LDS (mean = sum / max(count, 1))
    for (int i = tid; i < 16 * D_FEAT; i += 256) {
        int r = i >> 7;
        int c = i & (D_FEAT - 1);
        float denom = fmaxf(cnts[row0 + r], 1.0f);
        ldsA[r * LDS_STRIDE + c] = sums[(size_t)(row0 + r) * D_FEAT + c] / denom;
    }
    __syncthreads();

    const int wave = tid >> 5;        // 0..7 -> N-tile
    const int lane = tid & 31;
    const int m    = lane & 15;       // row within tile / col within N-tile
    const int kh   = lane >> 4;       // K half-select per ISA 16x4 f32 A layout
    const int n    = wave * 16 + m;   // output column
    const float* wrow = W + (size_t)n * D_FEAT;   // row n of W == column n of B

    v8f acc = {};
    #pragma unroll
    for (int s = 0; s < 32; ++s) {
        int k = 4 * s + 2 * kh;
        // A frag: lane holds mean[row0+m][k], mean[row0+m][k+1]
        v2f a = *(const v2f*)&ldsA[m * LDS_STRIDE + k];
        // B frag: lane holds W[n][k], W[n][k+1]  (B[k][n] = W[n][k])
        v2f b = *(const v2f*)(wrow + k);
        acc = __builtin_amdgcn_wmma_f32_16x16x4_f32(
                /*neg_a=*/false, a, /*neg_b=*/false, b,
                /*c_mod=*/(short)0, acc, /*reuse_a=*/false, /*reuse_b=*/false);
    }

    // C/D layout: VGPR v -> M = v + 8*kh, N = lane&15 ; coalesced stores
    float bv = bias[n];
    #pragma unroll
    for (int v = 0; v < 8; ++v) {
        int mr = v + 8 * kh;
        out[(size_t)(row0 + mr) * D_FEAT + n] = acc[v] + bv;
    }
}

// ---------------- launcher ----------------
extern "C" void kernel_launch(void* const* d_in, const int* in_sizes, int n_in,
                              void* d_out, int out_size, void* d_ws, size_t ws_size,
                              hipStream_t stream) {
    const float*     x  = (const float*)d_in[0];
    const long long* ei = (const long long*)d_in[1];   // int64 in reference
    const float*     W  = (const float*)d_in[4];
    const float*     b  = (const float*)d_in[5];
    float*           out = (float*)d_out;

    const int N = in_sizes[0] / D_FEAT;   // 50000 nodes
    const int E = in_sizes[1] / 2;        // 800000 edges

    float* sums = (float*)d_ws;                 // N * 128 floats
    float* cnts = sums + (size_t)N * D_FEAT;    // N floats

    int zn = N * D_FEAT + N;
    gcn_zero_kernel<<<(zn + 255) / 256, 256, 0, stream>>>(sums, zn);

    long long sthreads = (long long)E * 32;
    gcn_scatter_kernel<<<(int)((sthreads + 255) / 256), 256, 0, stream>>>(
        x, ei, sums, cnts, E);

    gcn_mean_gemm_kernel<<<(N + 15) / 16, 256, 0, stream>>>(sums, cnts, W, b, out);
}